// ImgGLUEncoder_36206574305378
// MI455X (gfx1250) — compile-verified
//
#include <hip/hip_runtime.h>
#include <math.h>

// ---------------------------------------------------------------------------
// Types for CDNA5 WMMA f32 16x16x4:  A = 16x4 f32 (2 VGPR), B = 4x16 f32
// (2 VGPR), C/D = 16x16 f32 (8 VGPR).  Wave32 only.
// ---------------------------------------------------------------------------
typedef __attribute__((ext_vector_type(2))) float v2f;
typedef __attribute__((ext_vector_type(8))) float v8f;

#define ACT_NONE 0
#define ACT_GELU 1
#define ACT_SIG  2

__device__ __forceinline__ float act_gelu(float x) {
    // exact erf gelu (torch F.gelu default / jax approximate=False)
    return 0.5f * x * (1.0f + erff(x * 0.70710678118654752440f));
}
__device__ __forceinline__ float act_sigmoid(float x) {
    return 1.0f / (1.0f + expf(-x));
}

// ---------------------------------------------------------------------------
// GEMM: OUT[M,N] = epilogue( A[M,K] @ W[K,N] + bias[N] )
//   epilogue: ACT in {none, gelu, sigmoid}; optional *mul (elementwise),
//   optional +res (residual).  Requires M%32==0, N%128==0, K%32==0.
// Block: 256 thr = 8 waves (2 M x 4 N); wave tile 16x32 (two 16x16 accs).
// ---------------------------------------------------------------------------
template <int ACT, bool HAS_MUL, bool HAS_RES>
__global__ __launch_bounds__(256) void gemm_wmma_f32(
    const float* __restrict__ A, const float* __restrict__ W,
    const float* __restrict__ bias, const float* __restrict__ mul,
    const float* __restrict__ res, float* __restrict__ out,
    int M, int N, int K)
{
    __shared__ float As[32][36];    // 32x32 tile, padded row stride (16B aligned)
    __shared__ float Bs[32][132];   // 32x128 tile, padded row stride (16B aligned)

    const int tid   = threadIdx.x;
    const int lane  = tid & 31;
    const int wave  = tid >> 5;
    const int waveM = wave >> 2;          // 0..1
    const int waveN = wave & 3;           // 0..3
    const int m0    = blockIdx.y * 32;
    const int n0    = blockIdx.x * 128;

    const int mrow  = lane & 15;          // M index inside 16 (A) / N index (B,C)
    const int khalf = lane >> 4;          // 0: K=0,1   1: K=2,3

    v8f acc0 = {};
    v8f acc1 = {};

    for (int kb = 0; kb < K; kb += 32) {
        // ---- stage A tile (32 rows x 32 k) : 4 floats / thread ----
        {
            const int row = tid >> 3;              // 0..31
            const int kc  = (tid & 7) << 2;        // 0..28
            const float4 v =
                *(const float4*)(A + (size_t)(m0 + row) * K + kb + kc);
            *(float4*)&As[row][kc] = v;
        }
        // ---- stage B tile (32 k x 128 n) : 16 floats / thread ----
        {
            const int col = (tid & 31) << 2;       // 0..124
            const int r0  = tid >> 5;              // 0..7
#pragma unroll
            for (int i = 0; i < 4; ++i) {
                const int row = r0 + i * 8;        // 0..31
                const float4 v =
                    *(const float4*)(W + (size_t)(kb + row) * N + n0 + col);
                *(float4*)&Bs[row][col] = v;
            }
        }
        __syncthreads();

        const int arow = waveM * 16 + mrow;
        const int nb   = waveN * 32;
#pragma unroll
        for (int kk = 0; kk < 32; kk += 4) {
            v2f a, b0, b1;
            a.x  = As[arow][kk + 2 * khalf + 0];
            a.y  = As[arow][kk + 2 * khalf + 1];
            b0.x = Bs[kk + 2 * khalf + 0][nb + mrow];
            b0.y = Bs[kk + 2 * khalf + 1][nb + mrow];
            b1.x = Bs[kk + 2 * khalf + 0][nb + 16 + mrow];
            b1.y = Bs[kk + 2 * khalf + 1][nb + 16 + mrow];
            acc0 = __builtin_amdgcn_wmma_f32_16x16x4_f32(
                false, a, false, b0, (short)0, acc0, false, false);
            acc1 = __builtin_amdgcn_wmma_f32_16x16x4_f32(
                false, a, false, b1, (short)0, acc1, false, false);
        }
        __syncthreads();
    }

    // ---- epilogue: C/D layout -> VGPR i holds (M = i + 8*khalf, N = lane&15)
    const int orow0 = m0 + waveM * 16 + khalf * 8;
    const int ocol0 = n0 + waveN * 32 + mrow;
#pragma unroll
    for (int i = 0; i < 8; ++i) {
        const int r = orow0 + i;
#pragma unroll
        for (int s = 0; s < 2; ++s) {
            const int c = ocol0 + s * 16;
            float v = (s == 0 ? acc0[i] : acc1[i]) + bias[c];
            if (ACT == ACT_GELU) v = act_gelu(v);
            if (ACT == ACT_SIG)  v = act_sigmoid(v);
            const size_t p = (size_t)r * N + c;
            if (HAS_MUL) v *= mul[p];
            if (HAS_RES) v += res[p];
            out[p] = v;
        }
    }
}

// ---------------------------------------------------------------------------
// RMSNorm: one wave per row of 512.  out = x * rsqrt(mean(x^2)+eps) * w
// ---------------------------------------------------------------------------
__global__ __launch_bounds__(256) void rms_kernel(
    const float* __restrict__ x, const float* __restrict__ w,
    float* __restrict__ out, int M)
{
    const int wave = threadIdx.x >> 5;
    const int lane = threadIdx.x & 31;
    const int row  = blockIdx.x * 8 + wave;
    if (row >= M) return;
    const float* xr = x + (size_t)row * 512;
    float v[16];
    float s = 0.f;
#pragma unroll
    for (int i = 0; i < 16; ++i) {
        v[i] = xr[lane + i * 32];
        s += v[i] * v[i];
    }
#pragma unroll
    for (int off = 16; off >= 1; off >>= 1) s += __shfl_xor(s, off, 32);
    const float scale = rsqrtf(s * (1.0f / 512.0f) + 1e-6f);
    float* o = out + (size_t)row * 512;
#pragma unroll
    for (int i = 0; i < 16; ++i) {
        const int c = lane + i * 32;
        o[c] = v[i] * scale * w[c];
    }
}

// ---------------------------------------------------------------------------
// CrossScan: xp[16,196,512] -> xs4[4][16][196][512] with the 4 scan orders
// ---------------------------------------------------------------------------
__global__ void crossscan_kernel(const float* __restrict__ xp,
                                 float* __restrict__ xs4)
{
    const int idx = blockIdx.x * blockDim.x + threadIdx.x;
    const int total = 4 * 16 * 196 * 512;
    if (idx >= total) return;
    const int d = idx & 511;
    int t = idx >> 9;
    const int l = t % 196; t /= 196;
    const int b = t % 16;
    const int k = t / 16;
    int src;
    if (k == 0)      src = l;
    else if (k == 1) src = (l % 14) * 14 + l / 14;
    else if (k == 2) src = 195 - l;
    else { const int j = 195 - l; src = (j % 14) * 14 + j / 14; }
    xs4[idx] = xp[((size_t)b * 196 + src) * 512 + d];
}

// ---------------------------------------------------------------------------
// CrossMerge: y4[4][16][196][512] -> out[16,196,512]
// ---------------------------------------------------------------------------
__global__ void crossmerge_kernel(const float* __restrict__ y4,
                                  float* __restrict__ out)
{
    const int idx = blockIdx.x * blockDim.x + threadIdx.x;
    const int total = 16 * 196 * 512;
    if (idx >= total) return;
    const int d = idx & 511;
    int t = idx >> 9;
    const int l = t % 196;
    const int b = t / 196;
    const int tl = (l % 14) * 14 + l / 14;
    const size_t SD = (size_t)16 * 196 * 512;
    const size_t rb = (size_t)b * 196;
    float v = y4[0 * SD + (rb + l) * 512 + d]
            + y4[2 * SD + (rb + (195 - l)) * 512 + d]
            + y4[1 * SD + (rb + tl) * 512 + d]
            + y4[3 * SD + (rb + (195 - tl)) * 512 + d];
    out[idx] = v;
}

// ---------------------------------------------------------------------------
// a/b for the recurrence (in place over r and ig buffers):
//   a = exp(-8*softplus(alpha[c]) * r),  b = sqrt(1-a^2)*hs*ig
// ---------------------------------------------------------------------------
__global__ void ab_kernel(const float* __restrict__ hs,
                          float* __restrict__ rA, float* __restrict__ igB,
                          const float* __restrict__ alpha, int total)
{
    const int idx = blockIdx.x * blockDim.x + threadIdx.x;
    if (idx >= total) return;
    const int c = idx & 511;
    const float sp = log1pf(expf(alpha[c]));
    const float a  = expf(-8.0f * sp * rA[idx]);
    const float b  = sqrtf(fmaxf(1.0f - a * a, 0.0f)) * hs[idx] * igB[idx];
    rA[idx]  = a;
    igB[idx] = b;
}

// ---------------------------------------------------------------------------
// Gated linear recurrence along L=196, fused with  X = out*gate + X
// One thread per (batch, channel): 16*512 = 8192 threads, coalesced over c.
// ---------------------------------------------------------------------------
__global__ void scan_kernel(const float* __restrict__ a,
                            const float* __restrict__ bb,
                            const float* __restrict__ gate,
                            float* __restrict__ X)
{
    const int idx = blockIdx.x * blockDim.x + threadIdx.x;
    if (idx >= 16 * 512) return;
    const int c = idx & 511;
    const int b = idx >> 9;
    const size_t base = (size_t)b * 196 * 512 + c;
    float o = 0.0f;
    for (int t = 0; t < 196; ++t) {
        const size_t p = base + (size_t)t * 512;
        o = a[p] * o + bb[p];
        X[p] = o * gate[p] + X[p];
    }
}

// ---------------------------------------------------------------------------
// Host launcher
// ---------------------------------------------------------------------------
extern "C" void kernel_launch(void* const* d_in, const int* in_sizes, int n_in,
                              void* d_out, int out_size, void* d_ws, size_t ws_size,
                              hipStream_t stream)
{
    (void)in_sizes; (void)n_in; (void)out_size; (void)ws_size;

    const float* x      = (const float*)d_in[0];
    const float* Wproj  = (const float*)d_in[1];
    const float* bproj  = (const float*)d_in[2];
    const float* alpha  = (const float*)d_in[3];
    const float* norm_w = (const float*)d_in[4];
    const float* Wfc    = (const float*)d_in[5];
    const float* bfc    = (const float*)d_in[6];
    const float* Wr     = (const float*)d_in[7];
    const float* br     = (const float*)d_in[8];
    const float* Wi     = (const float*)d_in[9];
    const float* bi     = (const float*)d_in[10];
    const float* Wg     = (const float*)d_in[11];
    const float* bg     = (const float*)d_in[12];
    const float* norm2w = (const float*)d_in[13];
    const float* Wmg    = (const float*)d_in[14];
    const float* bmg    = (const float*)d_in[15];
    const float* Wmf    = (const float*)d_in[16];
    const float* bmf    = (const float*)d_in[17];
    const float* Wout   = (const float*)d_in[18];
    const float* bout   = (const float*)d_in[19];
    float* out = (float*)d_out;

    const int M   = 16 * 196;   // 3136 tokens
    const int D   = 512;
    const int DIN = 1024;
    const int D3  = 1536;

    float* ws = (float*)d_ws;
    size_t off = 0;
    auto alloc = [&](size_t n) { float* p = ws + off; off += n; return p; };
    float* XP   = alloc((size_t)M * D);        // projected input
    float* XS4  = alloc((size_t)4 * M * D);    // 4 direction streams
    float* H    = alloc((size_t)M * D);        // rms output
    float* GATE = alloc((size_t)M * D);
    float* HS   = alloc((size_t)M * D);
    float* R    = alloc((size_t)M * D);        // r -> a (in place)
    float* IG   = alloc((size_t)M * D);        // ig -> b (in place)
    float* G2   = alloc((size_t)M * D3);
    float* F2   = alloc((size_t)M * D3);

    const dim3 blk(256);
    const int elemsMD  = M * D;
    const int gridMD   = (elemsMD + 255) / 256;
    const int grid4MD  = (4 * elemsMD + 255) / 256;
    const dim3 gD (D  / 128, M / 32);   // (4, 98)
    const dim3 gD3(D3 / 128, M / 32);   // (12, 98)

    // input projection + cross-scan
    gemm_wmma_f32<ACT_NONE, false, false><<<gD, blk, 0, stream>>>(
        x, Wproj, bproj, nullptr, nullptr, XP, M, D, DIN);
    crossscan_kernel<<<grid4MD, blk, 0, stream>>>(XP, XS4);

    for (int dir = 0; dir < 4; ++dir) {
        float* X = XS4 + (size_t)dir * M * D;
        for (int layer = 0; layer < 12; ++layer) {
            const size_t dl = (size_t)dir * 12 + layer;
            const float* p_nw   = norm_w + dl * D;
            const float* p_n2w  = norm2w + dl * D;
            const float* p_al   = alpha  + dl * D;
            const float* p_Wfc  = Wfc  + dl * D * D;
            const float* p_Wr   = Wr   + dl * D * D;
            const float* p_Wi   = Wi   + dl * D * D;
            const float* p_Wg   = Wg   + dl * D * D;
            const float* p_Wmg  = Wmg  + dl * D * D3;
            const float* p_Wmf  = Wmf  + dl * D * D3;
            const float* p_Wout = Wout + dl * D3 * D;
            const float* p_bfc  = bfc  + dl * D;
            const float* p_br   = br   + dl * D;
            const float* p_bi   = bi   + dl * D;
            const float* p_bg   = bg   + dl * D;
            const float* p_bmg  = bmg  + dl * D3;
            const float* p_bmf  = bmf  + dl * D3;
            const float* p_bout = bout + dl * D;

            // h = rms(X)*norm_w
            rms_kernel<<<M / 8, blk, 0, stream>>>(X, p_nw, H, M);
            // gate = gelu(h@Wg+bg); hs = h@Wfc+bfc
            gemm_wmma_f32<ACT_GELU, false, false><<<gD, blk, 0, stream>>>(
                H, p_Wg, p_bg, nullptr, nullptr, GATE, M, D, D);
            gemm_wmma_f32<ACT_NONE, false, false><<<gD, blk, 0, stream>>>(
                H, p_Wfc, p_bfc, nullptr, nullptr, HS, M, D, D);
            // r = sigmoid(hs@Wr+br); ig = sigmoid(hs@Wi+bi)
            gemm_wmma_f32<ACT_SIG, false, false><<<gD, blk, 0, stream>>>(
                HS, p_Wr, p_br, nullptr, nullptr, R, M, D, D);
            gemm_wmma_f32<ACT_SIG, false, false><<<gD, blk, 0, stream>>>(
                HS, p_Wi, p_bi, nullptr, nullptr, IG, M, D, D);
            // a,b then recurrence + gate + residual (X updated in place)
            ab_kernel<<<gridMD, blk, 0, stream>>>(HS, R, IG, p_al, elemsMD);
            scan_kernel<<<32, blk, 0, stream>>>(R, IG, GATE, X);
            // second half: GLU MLP with residual
            rms_kernel<<<M / 8, blk, 0, stream>>>(X, p_n2w, H, M);
            gemm_wmma_f32<ACT_GELU, false, false><<<gD3, blk, 0, stream>>>(
                H, p_Wmg, p_bmg, nullptr, nullptr, G2, M, D3, D);
            gemm_wmma_f32<ACT_NONE, true, false><<<gD3, blk, 0, stream>>>(
                H, p_Wmf, p_bmf, G2, nullptr, F2, M, D3, D);   // F2 = (h@Wmf+b)*G2
            gemm_wmma_f32<ACT_NONE, false, true><<<gD, blk, 0, stream>>>(
                F2, p_Wout, p_bout, nullptr, X, X, M, D, D3);  // X = F2@Wout+b+X
        }
    }

    crossmerge_kernel<<<gridMD, blk, 0, stream>>>(XS4, out);
}